// MultiHeadAttention_83150566851079
// MI455X (gfx1250) — compile-verified
//
#include <hip/hip_runtime.h>
#include <cmath>

// Problem constants (match reference)
#define B_    8
#define S_    1024
#define D_    1024
#define H_    16
#define DH_   64
#define MROWS (B_ * S_)      // 8192 token rows
#define LN_EPS 1e-5f
#define INV_SQRT_D 0.03125f  // 1/sqrt(1024)

typedef __attribute__((ext_vector_type(2))) float v2f;
typedef __attribute__((ext_vector_type(4))) float v4f;
typedef __attribute__((ext_vector_type(8))) float v8f;
typedef __attribute__((ext_vector_type(4))) unsigned int v4u;
typedef __attribute__((ext_vector_type(4))) int v4i;
typedef __attribute__((ext_vector_type(8))) int v8i;

// Use the Tensor Data Mover when the toolchain exposes it (arity differs:
// ROCm 7.2 / clang-22 = 5 args, amdgpu-toolchain / clang-23 = 6 args).
#if defined(__HIP_DEVICE_COMPILE__) &&                                        \
    __has_builtin(__builtin_amdgcn_tensor_load_to_lds) &&                     \
    __has_builtin(__builtin_amdgcn_s_wait_tensorcnt)
#define TDM_OK 1
#else
#define TDM_OK 0
#endif

static __device__ __forceinline__ v8f wmma_f32_16x16x4(v2f a, v2f b, v8f c) {
  // 8 args: (neg_a, A, neg_b, B, c_mod, C, reuse_a, reuse_b)
  return __builtin_amdgcn_wmma_f32_16x16x4_f32(false, a, false, b, (short)0, c,
                                               false, false);
}

static __device__ __forceinline__ v8f v8f_zero() {
  v8f z = {0.f, 0.f, 0.f, 0.f, 0.f, 0.f, 0.f, 0.f};
  return z;
}

#if TDM_OK
// Issue one TDM load: a [tile_rows x 32] f32 tile of a row-major
// [tensor_rows x D_] tensor into LDS, padded 32 data + 4 pad DWORDs per row
// (matches the stride-36 LDS layout used by the WMMA fragment loads).
static __device__ __forceinline__ void tdm_load_tile(const float* tile_gaddr,
                                                     unsigned lds_byte_off,
                                                     int tile_rows,
                                                     int tensor_rows) {
  const unsigned long long ga = (unsigned long long)(uintptr_t)tile_gaddr;
  v4u g0;
  g0.x = 1u;                                    // count=1, is_restore=0
  g0.y = lds_byte_off;                          // D#.lds_addr (bytes)
  g0.z = (unsigned)ga;                          // global_addr[31:0]
  g0.w = (unsigned)((ga >> 32) & 0x01FFFFFFull) // global_addr[56:32]
         | (2u << 30);                          // type = 2 ("image")

  v8i g1;
  // data_size=2 (4B), pad_enable, pad_interval=4 (pad every 32 DWORDs),
  // pad_amount=3 (4 DWORDs): reproduces the 36-float LDS row stride.
  g1[0] = (2 << 16) | (1 << 20) | (4 << 22) | (3 << 25);
  g1[1] = (int)((D_ & 0xFFFF) << 16);                       // tensor_dim0 lo16
  g1[2] = (int)((D_ >> 16) | ((tensor_rows & 0xFFFF) << 16)); // dim0 hi | dim1 lo
  g1[3] = (int)(((unsigned)tensor_rows >> 16) | (32u << 16)); // dim1 hi | tile_dim0=32
  g1[4] = tile_rows;                                        // tile_dim1 (tile_dim2=0)
  g1[5] = (int)D_;                                          // tensor_dim0_stride lo32
  g1[6] = 0;
  g1[7] = 0;

  v4i z4 = {0, 0, 0, 0};  // groups 2/3 unused for 2D tensors
#if defined(__clang_major__) && (__clang_major__ >= 23)
  v8i z8 = {0, 0, 0, 0, 0, 0, 0, 0};
  __builtin_amdgcn_tensor_load_to_lds(g0, g1, z4, z4, z8, 0);
#else
  __builtin_amdgcn_tensor_load_to_lds(g0, g1, z4, z4, 0);
#endif
}
#endif  // TDM_OK

// ---------------------------------------------------------------------------
// Kernel 1: Y = x @ W^T + bias, with the reference's (buggy) length mask.
// Grid: (D/64, MROWS/128), block 128 threads (4 waves). 128x64 tile per WG.
// Wave w owns rows [w*32, w*32+32) (2 M-subtiles) x all 64 cols (4 N-subtiles)
// -> 8 WMMA accumulators per wave, 6 ds_load_b64 per 8 WMMA.
// LDS is m-major, row stride 36 floats: fragments {k,k+1} are contiguous
// (single b64), and 36m mod 64 spreads half-waves over all 64 banks.
// TDM path: double-buffered tensor_load_to_lds overlaps DMA of chunk c+1
// with WMMA on chunk c (wave 0 drives the TDM; barriers publish the data).
// ---------------------------------------------------------------------------
__global__ __launch_bounds__(128) void qkv_gemm_kernel(
    const float* __restrict__ X,      // [MROWS, D]
    const float* __restrict__ W,      // [D, D] (row n holds W[n, :])
    const float* __restrict__ bias,   // [D]
    const int*   __restrict__ lengths,// [B]
    float* __restrict__ Y)            // [MROWS, D]
{
#if TDM_OK
  __shared__ float As[2][128][36];  // 36,864 B (double buffered)
  __shared__ float Bs[2][64][36];   // 18,432 B
#else
  __shared__ float As[1][128][36];  // 18,432 B
  __shared__ float Bs[1][64][36];   //  9,216 B
#endif

  const int tid  = threadIdx.x;
  const int lane = tid & 31;
  const int wv   = tid >> 5;          // wave 0..3
  const int m0g  = blockIdx.y * 128;
  const int n0g  = blockIdx.x * 64;
  const int lm   = lane & 15;         // fragment row/col index
  const int kb   = (lane >> 4) * 2;   // 0 or 2 (k sub-offset)

  v8f acc[2][4];
  #pragma unroll
  for (int mi = 0; mi < 2; ++mi)
    #pragma unroll
    for (int ni = 0; ni < 4; ++ni) acc[mi][ni] = v8f_zero();

  const int NC = D_ / 32;  // 32 K-chunks

#if TDM_OK
  // Prologue: kick off chunk 0 DMA.
  if (wv == 0) {
    tdm_load_tile(X + (size_t)m0g * D_, (unsigned)(uintptr_t)&As[0][0][0],
                  128, MROWS);
    tdm_load_tile(W + (size_t)n0g * D_, (unsigned)(uintptr_t)&Bs[0][0][0],
                  64, D_);
  }
#endif

  for (int c = 0; c < NC; ++c) {
    const int k0 = c * 32;
#if TDM_OK
    const int cur = c & 1;
    if (wv == 0) {
      if (c + 1 < NC) {
        // Issue next chunk into the other buffer, then retire current chunk
        // (2 newest loads may stay in flight -> DMA/compute overlap).
        tdm_load_tile(X + (size_t)m0g * D_ + (k0 + 32),
                      (unsigned)(uintptr_t)&As[cur ^ 1][0][0], 128, MROWS);
        tdm_load_tile(W + (size_t)n0g * D_ + (k0 + 32),
                      (unsigned)(uintptr_t)&Bs[cur ^ 1][0][0], 64, D_);
        __builtin_amdgcn_s_wait_tensorcnt(2);
      } else {
        __builtin_amdgcn_s_wait_tensorcnt(0);
      }
    }
    __syncthreads();  // chunk c data visible to all waves
#else
    const int cur = 0;
    __syncthreads();
    // Cooperative staging: coalesced v4f global loads -> contiguous b128 LDS
    #pragma unroll
    for (int it = 0; it < 8; ++it) {
      const int r = it * 16 + (tid >> 3);  // 0..127
      const int col = (tid & 7) * 4;       // 0..28 step 4
      const float* gx = X + (size_t)(m0g + r) * D_ + k0 + col;
      *(v4f*)&As[0][r][col] = *(const v4f*)gx;
      if (k0 + 32 < D_) __builtin_prefetch(gx + 32, 0, 3);
    }
    #pragma unroll
    for (int it = 0; it < 4; ++it) {
      const int r = it * 16 + (tid >> 3);  // 0..63
      const int col = (tid & 7) * 4;
      const float* gw = W + (size_t)(n0g + r) * D_ + k0 + col;
      *(v4f*)&Bs[0][r][col] = *(const v4f*)gw;
      if (k0 + 32 < D_) __builtin_prefetch(gw + 32, 0, 3);
    }
    __syncthreads();
#endif

    #pragma unroll
    for (int kk = 0; kk < 32; kk += 4) {
      v2f a0 = *(const v2f*)&As[cur][wv * 32 + lm][kk + kb];
      v2f a1 = *(const v2f*)&As[cur][wv * 32 + 16 + lm][kk + kb];
      #pragma unroll
      for (int ni = 0; ni < 4; ++ni) {
        v2f bf = *(const v2f*)&Bs[cur][ni * 16 + lm][kk + kb];
        acc[0][ni] = wmma_f32_16x16x4(a0, bf, acc[0][ni]);
        acc[1][ni] = wmma_f32_16x16x4(a1, bf, acc[1][ni]);
      }
    }
#if TDM_OK
    __syncthreads();  // all waves done reading buffer `cur` before its reuse
#endif
  }

  // Epilogue: bias + faithful mask (only flattened head-rows < B get zeroed:
  // batch 0, heads 0..7, positions >= lengths[head]).
  #pragma unroll
  for (int mi = 0; mi < 2; ++mi) {
    #pragma unroll
    for (int ni = 0; ni < 4; ++ni) {
      const int col  = n0g + ni * 16 + lm;
      const float bv = bias[col];
      const int head = col >> 6;  // col / DH
      #pragma unroll
      for (int r = 0; r < 8; ++r) {
        const int mrow = m0g + wv * 32 + mi * 16 + r + ((lane >> 4) << 3);
        float val = acc[mi][ni][r] + bv;
        const int bidx = mrow >> 10;   // / S_
        const int spos = mrow & (S_ - 1);
        if (bidx == 0 && head < B_ && spos >= lengths[head]) val = 0.f;
        Y[(size_t)mrow * D_ + col] = val;
      }
    }
  }
}

// ---------------------------------------------------------------------------
// Kernel 2: attention for one (head, 16-query-row) tile per workgroup.
// Grid: (S/16, B*H), block 128 threads (4 waves).
// Phase A: scores strip [16 x 1024] -> LDS (zero->1e-10, * 1/sqrt(D))
// Phase B: softmax stats (row max, sum of exp) in LDS, stride-8 interleaved
// Phase C: ctx = P @ V via WMMA, normalized by 1/rowsum at store
// PS = 1028: row stride ≡ 4 (mod 64) -> half-wave b64 fragment reads land on
// disjoint bank residues (mod 4): conflict-free.
// ---------------------------------------------------------------------------
#define PS 1028  // padded row stride for P (floats)

__global__ __launch_bounds__(128) void attn_kernel(
    const float* __restrict__ Q,   // [MROWS, D]
    const float* __restrict__ K,   // [MROWS, D]
    const float* __restrict__ V,   // [MROWS, D]
    float* __restrict__ Out)       // [MROWS, D] (ctx, merged heads)
{
  __shared__ float P[16 * PS];     // 65,792 B score strip
  __shared__ float red[16 * 8];
  __shared__ float rowmax[16];
  __shared__ float rinv[16];

  const int tid  = threadIdx.x;
  const int lane = tid & 31;
  const int wv   = tid >> 5;
  const int lm   = lane & 15;
  const int kb   = (lane >> 4) * 2;

  const int s0 = blockIdx.x * 16;
  const int bh = blockIdx.y;
  const int b  = bh >> 4;   // / H
  const int h  = bh & 15;
  const size_t hoff = (size_t)h * DH_;

  // Preload this lane's Q A-fragments for all 16 k-steps (dh = 0..63).
  const float* Qrow = Q + (size_t)(b * S_ + s0 + lm) * D_ + hoff;
  v2f qa[16];
  #pragma unroll
  for (int i = 0; i < 16; ++i)
    qa[i] = *(const v2f*)(Qrow + i * 4 + kb);

  // ---- Phase A: raw scores -> LDS ----
  for (int tt = wv; tt < S_ / 16; tt += 4) {
    const int t0 = tt * 16;
    const float* Krow = K + (size_t)(b * S_ + t0 + lm) * D_ + hoff;
    v8f acc = v8f_zero();
    #pragma unroll
    for (int i = 0; i < 16; ++i) {
      v2f kf = *(const v2f*)(Krow + i * 4 + kb);
      acc = wmma_f32_16x16x4(qa[i], kf, acc);
    }
    #pragma unroll
    for (int r = 0; r < 8; ++r) {
      const int m = r + ((lane >> 4) << 3);
      float v = acc[r];
      v = (v == 0.0f) ? 1e-10f : v;          // reference zero-replacement
      P[m * PS + t0 + lm] = v * INV_SQRT_D;  // then scale by 1/sqrt(D)
    }
  }
  __syncthreads();

  // ---- Phase B: softmax statistics (8 threads per row, stride-8 access) ----
  const int row = tid >> 3;
  const int seg = tid & 7;
  float* prow = P + row * PS;

  float mx = -3.0e38f;
  for (int j = 0; j < 128; ++j) mx = fmaxf(mx, prow[seg + j * 8]);
  red[row * 8 + seg] = mx;
  __syncthreads();
  if (seg == 0) {
    float m2 = red[row * 8];
    #pragma unroll
    for (int j = 1; j < 8; ++j) m2 = fmaxf(m2, red[row * 8 + j]);
    rowmax[row] = m2;
  }
  __syncthreads();

  const float rm = rowmax[row];
  float sm = 0.f;
  for (int j = 0; j < 128; ++j) {
    const int idx = seg + j * 8;
    float e = __expf(prow[idx] - rm);
    prow[idx] = e;
    sm += e;
  }
  red[row * 8 + seg] = sm;
  __syncthreads();
  if (seg == 0) {
    float s2 = 0.f;
    #pragma unroll
    for (int j = 0; j < 8; ++j) s2 += red[row * 8 + j];
    rinv[row] = 1.0f / s2;
  }
  __syncthreads();

  // ---- Phase C: ctx = P @ V (each wave owns one 16-wide dh slice) ----
  const int n0 = wv * 16;
  const float* Vcol = V + (size_t)(b * S_) * D_ + hoff + n0 + lm;
  v8f acc = v8f_zero();
  #pragma unroll 4
  for (int kk = 0; kk < S_; kk += 4) {
    v2f a = *(const v2f*)(P + lm * PS + kk + kb);
    v2f bf;
    bf.x = Vcol[(size_t)(kk + kb + 0) * D_];
    bf.y = Vcol[(size_t)(kk + kb + 1) * D_];
    acc = wmma_f32_16x16x4(a, bf, acc);
  }
  #pragma unroll
  for (int r = 0; r < 8; ++r) {
    const int m = r + ((lane >> 4) << 3);
    Out[(size_t)(b * S_ + s0 + m) * D_ + hoff + n0 + lm] = acc[r] * rinv[m];
  }
}

// ---------------------------------------------------------------------------
// Kernel 3: out = LayerNorm(ctx + x) * gamma + beta, in place on d_out.
// Grid: MROWS blocks of 256 threads; each thread handles 4 columns.
// ---------------------------------------------------------------------------
__global__ __launch_bounds__(256) void ln_kernel(
    const float* __restrict__ X, const float* __restrict__ gamma,
    const float* __restrict__ beta, float* __restrict__ Out)
{
  __shared__ float red[256];
  const int row = blockIdx.x;
  const int tid = threadIdx.x;
  const size_t base = (size_t)row * D_;

  v4f c  = *(const v4f*)(Out + base + tid * 4);
  v4f xv = *(const v4f*)(X + base + tid * 4);
  v4f v  = c + xv;  // residual

  red[tid] = v.x + v.y + v.z + v.w;
  __syncthreads();
  for (int st = 128; st > 0; st >>= 1) {
    if (tid < st) red[tid] += red[tid + st];
    __syncthreads();
  }
  const float mu = red[0] * (1.0f / D_);
  __syncthreads();

  const float d0 = v.x - mu, d1 = v.y - mu, d2 = v.z - mu, d3 = v.w - mu;
  red[tid] = d0 * d0 + d1 * d1 + d2 * d2 + d3 * d3;
  __syncthreads();
  for (int st = 128; st > 0; st >>= 1) {
    if (tid < st) red[tid] += red[tid + st];
    __syncthreads();
  }
  const float var = red[0] * (1.0f / D_);
  const float rs = rsqrtf(var + LN_EPS);

  v4f g  = *(const v4f*)(gamma + tid * 4);
  v4f be = *(const v4f*)(beta + tid * 4);
  v4f o;
  o.x = d0 * rs * g.x + be.x;
  o.y = d1 * rs * g.y + be.y;
  o.z = d2 * rs * g.z + be.z;
  o.w = d3 * rs * g.w + be.w;
  *(v4f*)(Out + base + tid * 4) = o;
}

// ---------------------------------------------------------------------------
extern "C" void kernel_launch(void* const* d_in, const int* in_sizes, int n_in,
                              void* d_out, int out_size, void* d_ws,
                              size_t ws_size, hipStream_t stream) {
  (void)in_sizes; (void)n_in; (void)out_size; (void)ws_size;
  const float* x       = (const float*)d_in[0];
  const int*   lengths = (const int*)d_in[1];
  const float* Wq = (const float*)d_in[2];
  const float* bq = (const float*)d_in[3];
  const float* Wk = (const float*)d_in[4];
  const float* bk = (const float*)d_in[5];
  const float* Wv = (const float*)d_in[6];
  const float* bv = (const float*)d_in[7];
  const float* gamma = (const float*)d_in[8];
  const float* beta  = (const float*)d_in[9];
  float* out = (float*)d_out;

  // Workspace: Q,K,V each MROWS*D f32 (32 MB each, 96 MB total).
  float* Qw = (float*)d_ws;
  float* Kw = Qw + (size_t)MROWS * D_;
  float* Vw = Kw + (size_t)MROWS * D_;

  dim3 g1(D_ / 64, MROWS / 128);
  qkv_gemm_kernel<<<g1, 128, 0, stream>>>(x, Wq, bq, lengths, Qw);
  qkv_gemm_kernel<<<g1, 128, 0, stream>>>(x, Wk, bk, lengths, Kw);
  qkv_gemm_kernel<<<g1, 128, 0, stream>>>(x, Wv, bv, lengths, Vw);

  attn_kernel<<<dim3(S_ / 16, B_ * H_), 128, 0, stream>>>(Qw, Kw, Vw, out);

  ln_kernel<<<dim3(MROWS), 256, 0, stream>>>(x, gamma, beta, out);
}